// MRConv_1632087572827
// MI455X (gfx1250) — compile-verified
//
#include <hip/hip_runtime.h>

// MRConv on gfx1250 (MI455X): gather/max-relative feature build + fp32 WMMA GEMM.
//
// Sizes: B=4, C=192, N=10000, K=16, OUT=384, 2C=384, COLS=B*N=40000.
// Workspace layout (floats):
//   xT   : [B][N][C]        7,680,000
//   feat : [2C][COLS]      15,360,000   (k=2c -> x, k=2c+1 -> max_k(x_j - x_i))
//   Wp   : WMMA-A-fragment-ordered W, 24*96*32*2 = 147,456
// Total = 92,749,824 bytes.

#define Bq   4
#define Cq   192
#define Nq   10000
#define Kq   16
#define OUTq 384
#define C2q  384
#define COLS 40000
#define KSTEPS 96           // C2q / 4
#define MTILES 24           // OUTq / 16
#define NTILES 2500         // COLS / 16

typedef __attribute__((ext_vector_type(2))) float v2f;
typedef __attribute__((ext_vector_type(8))) float v8f;

// ---------------------------------------------------------------------------
// Kernel A: xT[b][n][c] = x[b][c][n]  (LDS tiled transpose)
//           feat[2c][b*N+n] = x[b][c][n] (even rows, straight coalesced copy)
// ---------------------------------------------------------------------------
__global__ void mrconv_transpose(const float* __restrict__ x,
                                 float* __restrict__ xT,
                                 float* __restrict__ feat) {
    __shared__ float tile[32][33];
    const int n0 = blockIdx.x * 32;
    const int c0 = blockIdx.y * 32;
    const int b  = blockIdx.z;
    const int tx = threadIdx.x, ty = threadIdx.y;

    #pragma unroll
    for (int i = 0; i < 4; ++i) {
        const int c = c0 + ty + 8 * i;
        const int n = n0 + tx;
        if (n < Nq) {
            const float v = x[((long)b * Cq + c) * Nq + n];
            tile[ty + 8 * i][tx] = v;
            feat[(long)(2 * c) * COLS + b * Nq + n] = v;   // even feat row
        }
    }
    __syncthreads();
    #pragma unroll
    for (int i = 0; i < 4; ++i) {
        const int n = n0 + ty + 8 * i;
        const int c = c0 + tx;
        if (n < Nq) {
            xT[((long)b * Nq + n) * Cq + c] = tile[tx][ty + 8 * i];
        }
    }
}

// ---------------------------------------------------------------------------
// Kernel B: feat[2c+1][b*N+n] = max_k( xT[b][e0[b,n,k]][c] - xT[b][e1[b,n,k]][c] )
// Workgroup: 256 threads (8 waves) handles one batch b and 32 nodes.
// Lanes span a 32-channel chunk -> 128B coalesced gathers from L2-resident xT.
// Results staged in LDS, written out as coalesced 128B row segments.
// ---------------------------------------------------------------------------
__global__ void mrconv_gather_max(const float* __restrict__ xT,
                                  const int* __restrict__ eidx,
                                  float* __restrict__ feat) {
    __shared__ float sh[Cq * 33];       // [c][node] stride 33 -> conflict-free
    const int b    = blockIdx.y;
    const int n0   = blockIdx.x * 32;
    const int wave = threadIdx.x >> 5;
    const int lane = threadIdx.x & 31;

    #pragma unroll
    for (int u = 0; u < 4; ++u) {
        const int nd = wave * 4 + u;
        const int n  = n0 + nd;
        if (n < Nq) {
            const long ebase = ((long)b * Nq + n) * Kq;
            int jv = 0, iv = 0;
            if (lane < Kq) {
                jv = eidx[ebase + lane];                      // edge_index[0] (neighbors)
                iv = eidx[(long)Bq * Nq * Kq + ebase + lane]; // edge_index[1] (centers)
            }
            const float* xb = xT + (long)b * Nq * Cq;
            #pragma unroll
            for (int chunk = 0; chunk < 6; ++chunk) {
                const int c = chunk * 32 + lane;
                float m = -__builtin_inff();
                #pragma unroll
                for (int k = 0; k < Kq; ++k) {
                    const int j = __shfl(jv, k, 32);
                    const int i = __shfl(iv, k, 32);
                    m = fmaxf(m, xb[(long)j * Cq + c] - xb[(long)i * Cq + c]);
                }
                sh[c * 33 + nd] = m;
            }
        }
    }
    __syncthreads();
    // write odd feat rows: each 32-lane group emits one 128B row segment
    #pragma unroll
    for (int r = 0; r < 24; ++r) {
        const int c  = r * 8 + wave;
        const int nd = lane;
        const int n  = n0 + nd;
        if (n < Nq)
            feat[(long)(2 * c + 1) * COLS + b * Nq + n] = sh[c * 33 + nd];
    }
}

// ---------------------------------------------------------------------------
// Kernel C: repack W[OUT][2C] into WMMA 32-bit A-matrix 16x4 fragment order.
// ISA 7.12.2: A[m,k] -> lane = m + 16*(k>>1), vgpr = k&1.
// Wp[((mt*96+ks)*32 + lane)*2 + j] = W[mt*16 + (lane&15)][4*ks + 2*(lane>>4) + j]
// ---------------------------------------------------------------------------
__global__ void mrconv_prep_w(const float* __restrict__ W, float* __restrict__ Wp) {
    const int t = blockIdx.x * 256 + threadIdx.x;
    if (t >= MTILES * KSTEPS * 64) return;
    const int j    = t & 1;
    const int lane = (t >> 1) & 31;
    const int ks   = (t >> 6) % KSTEPS;
    const int mt   = t / (KSTEPS * 64);
    const int m = lane & 15;
    const int k = 4 * ks + 2 * (lane >> 4) + j;
    Wp[t] = W[(mt * 16 + m) * C2q + k];
}

// ---------------------------------------------------------------------------
// Kernel D: out[b][o][n] = relu( sum_k W[o][k]*feat[k][col] + bias[o] )
// Exact fp32 via V_WMMA_F32_16X16X4_F32. 4 waves/block; wave owns 6 M-tiles
// and TWO 16-col tiles: each A-fragment (the expensive operand, re-read from
// L2 by every block) is reused across both column tiles -> A traffic halved,
// issue ratio 10 loads : 12 WMMA per K-step.
// B 4x16 layout (mirror of A): lane = n + 16*(k>>1), vgpr = k&1.
// C/D layout: vgpr r -> M = r + 8*(lane>>4), N = lane&15.
// ---------------------------------------------------------------------------
__global__ void mrconv_gemm(const float* __restrict__ feat,
                            const float* __restrict__ Wp,
                            const float* __restrict__ bias,
                            float* __restrict__ out) {
    const int wave = threadIdx.x >> 5;
    const int lane = threadIdx.x & 31;
    const int col0 = blockIdx.x * 32;         // two adjacent 16-col tiles
    const int cA   = col0 + (lane & 15);
    const int kb   = lane >> 4;               // lane-half selects K pair
    const v2f* __restrict__ Wp2 = (const v2f*)Wp;

    const v8f zero = {0.f, 0.f, 0.f, 0.f, 0.f, 0.f, 0.f, 0.f};
    v8f acc[12];                              // [coltile][mtile] -> [t*6 + i]
    #pragma unroll
    for (int i = 0; i < 12; ++i) acc[i] = zero;

    for (int ks = 0; ks < KSTEPS; ++ks) {
        const int kbase = 4 * ks + 2 * kb;
        const float* frow0 = feat + (long)kbase * COLS;
        const float* frow1 = feat + (long)(kbase + 1) * COLS;
        v2f bf0, bf1;
        bf0.x = frow0[cA];      bf0.y = frow1[cA];
        bf1.x = frow0[cA + 16]; bf1.y = frow1[cA + 16];
        #pragma unroll
        for (int i = 0; i < 6; ++i) {
            const int mt = wave * 6 + i;
            const v2f a = Wp2[((mt * KSTEPS + ks) << 5) + lane];
            acc[i]     = __builtin_amdgcn_wmma_f32_16x16x4_f32(
                false, a, false, bf0, (short)0, acc[i],     false, false);
            acc[6 + i] = __builtin_amdgcn_wmma_f32_16x16x4_f32(
                false, a, false, bf1, (short)0, acc[6 + i], false, false);
        }
    }

    #pragma unroll
    for (int t = 0; t < 2; ++t) {
        const int col = col0 + t * 16 + (lane & 15);
        const int b = col / Nq;
        const int n = col - b * Nq;           // 16-col tile never straddles b
        #pragma unroll
        for (int i = 0; i < 6; ++i) {
            const int mtBase = (wave * 6 + i) * 16;
            #pragma unroll
            for (int r = 0; r < 8; ++r) {
                const int o = mtBase + r + 8 * kb;
                float v = acc[t * 6 + i][r] + bias[o];
                v = v > 0.f ? v : 0.f;
                out[((long)(b * OUTq + o)) * Nq + n] = v;
            }
        }
    }
}

// ---------------------------------------------------------------------------
extern "C" void kernel_launch(void* const* d_in, const int* in_sizes, int n_in,
                              void* d_out, int out_size, void* d_ws, size_t ws_size,
                              hipStream_t stream) {
    const float* x    = (const float*)d_in[0];   // [B,C,N,1] fp32
    const int*   eidx = (const int*)d_in[1];     // [2,B,N,K] int32
    const float* W    = (const float*)d_in[2];   // [OUT,2C] fp32
    const float* bias = (const float*)d_in[3];   // [OUT] fp32
    float* out = (float*)d_out;                  // [B,OUT,N,1,1] fp32

    float* ws   = (float*)d_ws;
    float* xT   = ws;                                      // 7,680,000 floats
    float* feat = ws + (long)Bq * Nq * Cq;                 // 15,360,000 floats
    float* Wp   = feat + (long)C2q * COLS;                 // 147,456 floats

    mrconv_transpose<<<dim3(313, 6, Bq), dim3(32, 8), 0, stream>>>(x, xT, feat);
    mrconv_gather_max<<<dim3(313, Bq), 256, 0, stream>>>(xT, eidx, feat);
    mrconv_prep_w<<<(MTILES * KSTEPS * 64 + 255) / 256, 256, 0, stream>>>(W, Wp);
    mrconv_gemm<<<NTILES / 2, 128, 0, stream>>>(feat, Wp, bias, out);
}